// RelTemporalEncoding_69956427317268
// MI455X (gfx1250) — compile-verified
//
#include <hip/hip_runtime.h>

// RelTemporalEncoding fused kernel for gfx1250 (MI455X, wave32, WMMA).
//
// out[n] = (sum_k wts[k] * emb[t[n,k]]) @ W^T + b      (wts sum to exactly 1.0)
//
// Phase A: vectorized (b64) gather of the L2-resident sinusoidal table into a
//          128x64 Z tile in LDS.
// Phase B: per-wave 16x64 tile via V_WMMA_F32_16X16X4_F32; B-fragments are
//          contiguous b64 LDS loads from row-major W (stride 68 => bank-free).
// Phase C: accumulators (+bias) written back to LDS, then streamed out as
//          fully-coalesced float4 (b128) global stores.

typedef float v2f __attribute__((ext_vector_type(2)));
typedef float v8f __attribute__((ext_vector_type(8)));

#define N_HID   62
#define KPAD    64
#define WSTRIDE 68   // row stride (floats) for W in LDS: conflict-free b64 B-frag reads
#define ZSTRIDE 68   // row stride (floats) for Z in LDS: conflict-free b64 A-frag reads
#define ROWS_PER_BLOCK 128
#define THREADS 256  // 8 waves of 32
#define PAIRS   (N_HID / 2)   // 31 b64 gathers per row

__global__ __launch_bounds__(THREADS, 2)
void rte_fused_wmma_kernel(const int*   __restrict__ t,     // [N,3] int32
                           const float* __restrict__ emb,   // [3000,62]
                           const float* __restrict__ W,     // [62,62]
                           const float* __restrict__ bias,  // [62]
                           float*       __restrict__ out,   // [N,62]
                           int n_rows) {
    __shared__ __align__(16) float sW[KPAD * WSTRIDE];           // row-major W, padded
    __shared__ __align__(16) float sZ[ROWS_PER_BLOCK * ZSTRIDE]; // Z tile / result tile
    __shared__ int   sT[ROWS_PER_BLOCK * 3];
    __shared__ float sB[KPAD];

    const int tid     = threadIdx.x;
    const int rowBase = blockIdx.x * ROWS_PER_BLOCK;

    // ---- Stage W row-major (zero-padded to 64x64) into LDS ----
    for (int i = tid; i < KPAD * KPAD; i += THREADS) {
        const int o = i >> 6;      // output-col index 0..63
        const int d = i & 63;      // K index 0..63
        float v = 0.0f;
        if (o < N_HID && d < N_HID) v = W[o * N_HID + d];
        sW[o * WSTRIDE + d] = v;
    }
    // ---- Stage bias (zero-padded) ----
    if (tid < KPAD) sB[tid] = (tid < N_HID) ? bias[tid] : 0.0f;
    // ---- Stage this block's t values ----
    for (int i = tid; i < ROWS_PER_BLOCK * 3; i += THREADS) {
        const int r = rowBase + i / 3;
        sT[i] = (r < n_rows) ? t[(size_t)rowBase * 3 + i] : 0;
    }
    __syncthreads();

    // ---- Phase A: Z[r][d] = wh*emb[t0][d] + wm*emb[t1][d] + ws*emb[t2][d] ----
    // Pairwise (b64) gathers: d even => 8B-aligned in both emb and sZ.
    const float wh = 3600.0f / 3661.0f;
    const float wm =   60.0f / 3661.0f;
    const float ws =    1.0f / 3661.0f;
    for (int p = tid; p < ROWS_PER_BLOCK * PAIRS; p += THREADS) {
        const int r = p / PAIRS;
        const int d = (p - r * PAIRS) * 2;
        v2f z = {0.0f, 0.0f};
        if (rowBase + r < n_rows) {
            const int t0 = sT[r * 3 + 0];
            const int t1 = sT[r * 3 + 1];
            const int t2 = sT[r * 3 + 2];
            const v2f e0 = *(const v2f*)(emb + t0 * N_HID + d);
            const v2f e1 = *(const v2f*)(emb + t1 * N_HID + d);
            const v2f e2 = *(const v2f*)(emb + t2 * N_HID + d);
            z.x = fmaf(wh, e0.x, fmaf(wm, e1.x, ws * e2.x));
            z.y = fmaf(wh, e0.y, fmaf(wm, e1.y, ws * e2.y));
        }
        *(v2f*)(&sZ[r * ZSTRIDE + d]) = z;
    }
    // zero K padding (dims 62, 63)
    for (int r = tid; r < ROWS_PER_BLOCK; r += THREADS) {
        *(v2f*)(&sZ[r * ZSTRIDE + N_HID]) = (v2f){0.0f, 0.0f};
    }
    __syncthreads();

    // ---- Phase B: per-wave 16x64 GEMM tile with V_WMMA_F32_16X16X4_F32 ----
    const int wave  = tid >> 5;      // 0..7 -> row tile
    const int lane  = tid & 31;
    const int lhalf = lane & 15;     // M (A) / N (B,D) position
    const int hi    = lane >> 4;     // 0: K0/K1 half, 1: K2/K3 half
    const int ztile = wave * 16;

    v8f acc0 = {}, acc1 = {}, acc2 = {}, acc3 = {};

    // A-frag source row for this lane (fixed); K advances along the row.
    const float* zrow = &sZ[(ztile + lhalf) * ZSTRIDE + hi * 2];
    // B-frag source rows: one per column tile, contiguous (k,k+1) pairs.
    const float* wr0 = &sW[(lhalf +  0) * WSTRIDE + hi * 2];
    const float* wr1 = &sW[(lhalf + 16) * WSTRIDE + hi * 2];
    const float* wr2 = &sW[(lhalf + 32) * WSTRIDE + hi * 2];
    const float* wr3 = &sW[(lhalf + 48) * WSTRIDE + hi * 2];

#pragma unroll
    for (int kk = 0; kk < KPAD / 4; ++kk) {
        const v2f a  = *(const v2f*)(zrow + kk * 4);   // A[M=lhalf][K..K+1]
        const v2f b0 = *(const v2f*)(wr0 + kk * 4);    // B[K..K+1][N=lhalf]
        const v2f b1 = *(const v2f*)(wr1 + kk * 4);
        const v2f b2 = *(const v2f*)(wr2 + kk * 4);
        const v2f b3 = *(const v2f*)(wr3 + kk * 4);

        acc0 = __builtin_amdgcn_wmma_f32_16x16x4_f32(false, a, false, b0, (short)0, acc0, false, false);
        acc1 = __builtin_amdgcn_wmma_f32_16x16x4_f32(false, a, false, b1, (short)0, acc1, false, false);
        acc2 = __builtin_amdgcn_wmma_f32_16x16x4_f32(false, a, false, b2, (short)0, acc2, false, false);
        acc3 = __builtin_amdgcn_wmma_f32_16x16x4_f32(false, a, false, b3, (short)0, acc3, false, false);
    }

    // ---- Write accumulators (+bias) back to this wave's own Z rows ----
    // D layout: VGPR v holds row (v + 8*hi), col = 16*ct + lhalf.
    {
        v8f accs[4] = { acc0, acc1, acc2, acc3 };
#pragma unroll
        for (int ct = 0; ct < 4; ++ct) {
            const int col = ct * 16 + lhalf;
            const float bv = sB[col];
            const v8f a = accs[ct];
#pragma unroll
            for (int v = 0; v < 8; ++v) {
                sZ[(ztile + hi * 8 + v) * ZSTRIDE + col] = a[v] + bv;
            }
        }
    }
    __syncthreads();

    // ---- Phase C: coalesced copy-out (drop the col 62/63 padding) ----
    if (rowBase + ROWS_PER_BLOCK <= n_rows) {
        // Full tile: 128*62 contiguous floats, 16B-aligned -> b128 stores.
        float* outBase = out + (size_t)rowBase * N_HID;
        for (int q = tid; q < (ROWS_PER_BLOCK * N_HID) / 4; q += THREADS) {
            const int idx = q * 4;
            float4 v;
            {
                const int i0 = idx + 0, r = i0 / N_HID, d = i0 - r * N_HID;
                v.x = sZ[r * ZSTRIDE + d];
            }
            {
                const int i1 = idx + 1, r = i1 / N_HID, d = i1 - r * N_HID;
                v.y = sZ[r * ZSTRIDE + d];
            }
            {
                const int i2 = idx + 2, r = i2 / N_HID, d = i2 - r * N_HID;
                v.z = sZ[r * ZSTRIDE + d];
            }
            {
                const int i3 = idx + 3, r = i3 / N_HID, d = i3 - r * N_HID;
                v.w = sZ[r * ZSTRIDE + d];
            }
            *(float4*)(outBase + idx) = v;
        }
    } else {
        // Tail tile: guarded scalar stores.
        for (int e = tid; e < ROWS_PER_BLOCK * N_HID; e += THREADS) {
            const int r = e / N_HID;
            const int d = e - r * N_HID;
            if (rowBase + r < n_rows)
                out[(size_t)(rowBase + r) * N_HID + d] = sZ[r * ZSTRIDE + d];
        }
    }
}

extern "C" void kernel_launch(void* const* d_in, const int* in_sizes, int n_in,
                              void* d_out, int out_size, void* d_ws, size_t ws_size,
                              hipStream_t stream) {
    const int*   t    = (const int*)  d_in[0];   // [N,3]
    const float* emb  = (const float*)d_in[1];   // [3000,62]
    const float* W    = (const float*)d_in[2];   // [62,62]
    const float* bias = (const float*)d_in[3];   // [62]
    float*       out  = (float*)d_out;           // [N,62]

    const int n_rows = in_sizes[0] / 3;
    const int blocks = (n_rows + ROWS_PER_BLOCK - 1) / ROWS_PER_BLOCK;

    rte_fused_wmma_kernel<<<blocks, THREADS, 0, stream>>>(t, emb, W, bias, out, n_rows);
}